// BerhuLoss_28793460752550
// MI455X (gfx1250) — compile-verified
//
#include <hip/hip_runtime.h>

// ---------------------------------------------------------------------------
// BerHu loss, shapes fixed by the reference: predictions/targets (64,1,480,640)
// fp32. Memory-bound two-pass reduction; final 64-way reduction uses
// v_wmma_f32_16x16x4_f32 on the matrix pipe.
// ---------------------------------------------------------------------------

typedef __attribute__((ext_vector_type(2))) float v2f;
typedef __attribute__((ext_vector_type(8))) float v8f;

constexpr int N_SAMPLES         = 64;
constexpr int ELEMS_PER_SAMPLE  = 480 * 640;          // 307200
constexpr int VEC4_PER_SAMPLE   = ELEMS_PER_SAMPLE/4; // 76800 float4
constexpr int THREADS           = 256;                // 8 wave32 per block
constexpr int BLOCKS_PER_SAMPLE = 128;
constexpr int WAVES_PER_BLOCK   = THREADS / 32;

// ---------------- workspace layout (offsets in bytes) ----------------------
//   [0, 256)                : unsigned int ws_max[64]   (bit-pattern maxima)
//   [256, 256 + 64*128*4)   : float ws_part[64][128]    (per-block sums)
// ---------------------------------------------------------------------------

__global__ void berhu_init_kernel(unsigned int* __restrict__ ws_max) {
    int t = threadIdx.x;
    if (t < N_SAMPLES) ws_max[t] = 0u;   // 0.0f bit pattern
}

__device__ __forceinline__ float wave_reduce_max(float v) {
    #pragma unroll
    for (int off = 16; off > 0; off >>= 1)
        v = fmaxf(v, __shfl_xor(v, off, 32));
    return v;
}

__device__ __forceinline__ float wave_reduce_sum(float v) {
    #pragma unroll
    for (int off = 16; off > 0; off >>= 1)
        v += __shfl_xor(v, off, 32);
    return v;
}

// Pass 1: per-sample max of |pred - targ|. atomicMax on uint bits is exact and
// order-independent (|d| >= 0), so the result is deterministic.
__global__ void berhu_max_kernel(const float4* __restrict__ pred,
                                 const float4* __restrict__ targ,
                                 unsigned int* __restrict__ ws_max) {
    const int sample = blockIdx.y;
    const float4* __restrict__ p = pred + (size_t)sample * VEC4_PER_SAMPLE;
    const float4* __restrict__ t = targ + (size_t)sample * VEC4_PER_SAMPLE;

    const int stride = gridDim.x * blockDim.x;
    float m = 0.0f;
    for (int i = blockIdx.x * blockDim.x + threadIdx.x; i < VEC4_PER_SAMPLE; i += stride) {
        if (i + stride < VEC4_PER_SAMPLE) {
            __builtin_prefetch(p + i + stride, 0, 1);   // global_prefetch_b8
            __builtin_prefetch(t + i + stride, 0, 1);
        }
        float4 a = p[i];
        float4 b = t[i];
        m = fmaxf(m, fabsf(a.x - b.x));
        m = fmaxf(m, fabsf(a.y - b.y));
        m = fmaxf(m, fabsf(a.z - b.z));
        m = fmaxf(m, fabsf(a.w - b.w));
    }

    m = wave_reduce_max(m);

    __shared__ float smax[WAVES_PER_BLOCK];
    const int lane = threadIdx.x & 31;
    const int wave = threadIdx.x >> 5;
    if (lane == 0) smax[wave] = m;
    __syncthreads();
    if (threadIdx.x == 0) {
        float bm = smax[0];
        #pragma unroll
        for (int w = 1; w < WAVES_PER_BLOCK; ++w) bm = fmaxf(bm, smax[w]);
        atomicMax(&ws_max[sample], __float_as_uint(bm));
    }
}

// Pass 2: per-block BerHu partial sums, written to fixed slots (no float
// atomics -> bitwise deterministic). Inputs (157 MB total) fit in 192 MB L2,
// so this pass should be largely L2-resident.
__global__ void berhu_sum_kernel(const float4* __restrict__ pred,
                                 const float4* __restrict__ targ,
                                 const unsigned int* __restrict__ ws_max,
                                 float* __restrict__ ws_part) {
    const int sample = blockIdx.y;
    const float4* __restrict__ p = pred + (size_t)sample * VEC4_PER_SAMPLE;
    const float4* __restrict__ t = targ + (size_t)sample * VEC4_PER_SAMPLE;

    const float c          = __uint_as_float(ws_max[sample]) * 0.2f; // max/5
    const float csq        = c * c;
    const float half_inv_c = 0.5f / fmaxf(c, 1e-30f);

    const int stride = gridDim.x * blockDim.x;
    float s = 0.0f;
    for (int i = blockIdx.x * blockDim.x + threadIdx.x; i < VEC4_PER_SAMPLE; i += stride) {
        if (i + stride < VEC4_PER_SAMPLE) {
            __builtin_prefetch(p + i + stride, 0, 1);
            __builtin_prefetch(t + i + stride, 0, 1);
        }
        float4 a = p[i];
        float4 b = t[i];
        float d0 = a.x - b.x, d1 = a.y - b.y, d2 = a.z - b.z, d3 = a.w - b.w;
        float a0 = fabsf(d0), a1 = fabsf(d1), a2 = fabsf(d2), a3 = fabsf(d3);
        s += (a0 <= c) ? a0 : (d0 * d0 + csq) * half_inv_c;
        s += (a1 <= c) ? a1 : (d1 * d1 + csq) * half_inv_c;
        s += (a2 <= c) ? a2 : (d2 * d2 + csq) * half_inv_c;
        s += (a3 <= c) ? a3 : (d3 * d3 + csq) * half_inv_c;
    }

    s = wave_reduce_sum(s);

    __shared__ float ssum[WAVES_PER_BLOCK];
    const int lane = threadIdx.x & 31;
    const int wave = threadIdx.x >> 5;
    if (lane == 0) ssum[wave] = s;
    __syncthreads();
    if (threadIdx.x == 0) {
        float bs = ssum[0];
        #pragma unroll
        for (int w = 1; w < WAVES_PER_BLOCK; ++w) bs += ssum[w];
        ws_part[(size_t)sample * BLOCKS_PER_SAMPLE + blockIdx.x] = bs;
    }
}

// Final: fixed-order reduction of the 128 partials per sample, then the 64
// per-sample means are summed with one V_WMMA_F32_16X16X4_F32:
//   A (16x4) holds the 64 means (lane L, vgpr v -> elem (L%16)*4+(L/16)*2+v),
//   B = all-ones 4x16 (layout irrelevant), C = 0.
//   D[m][n] = sum_k A[m][k]; rows 0-7 in lanes 0-15 (VGPR 0-7), rows 8-15 in
//   lanes 16-31. Total = per-lane sum of 8 D regs + shfl_xor(...,16).
__global__ void berhu_final_kernel(const float* __restrict__ ws_part,
                                   float* __restrict__ out) {
    __shared__ float smean[N_SAMPLES];
    const int t = threadIdx.x;
    if (t < N_SAMPLES) {
        const float* __restrict__ p = ws_part + (size_t)t * BLOCKS_PER_SAMPLE;
        float s = 0.0f;
        #pragma unroll 4
        for (int b = 0; b < BLOCKS_PER_SAMPLE; ++b) s += p[b];
        smean[t] = s * (1.0f / (float)ELEMS_PER_SAMPLE);
    }
    __syncthreads();

    if (t < 32) {                       // wave 0 only: full EXEC for WMMA
        const int m  = t & 15;
        const int kb = (t >> 4) * 2;    // lanes 0-15 -> K=0,1 ; lanes 16-31 -> K=2,3
        v2f a;
        a.x = smean[m * 4 + kb + 0];
        a.y = smean[m * 4 + kb + 1];
        v2f ones;
        ones.x = 1.0f;
        ones.y = 1.0f;
        v8f c0 = {};
        // 8 args: (neg_a, A, neg_b, B, c_mod, C, reuse_a, reuse_b)
        v8f d = __builtin_amdgcn_wmma_f32_16x16x4_f32(
            false, a, false, ones, (short)0, c0, false, false);
        float s8 = d[0] + d[1] + d[2] + d[3] + d[4] + d[5] + d[6] + d[7];
        float tot = s8 + __shfl_xor(s8, 16, 32);
        if (t == 0) out[0] = tot * (1.0f / (float)N_SAMPLES);
    }
}

extern "C" void kernel_launch(void* const* d_in, const int* in_sizes, int n_in,
                              void* d_out, int out_size, void* d_ws, size_t ws_size,
                              hipStream_t stream) {
    const float4* pred = (const float4*)d_in[0];
    const float4* targ = (const float4*)d_in[1];
    unsigned int* ws_max  = (unsigned int*)d_ws;
    float*        ws_part = (float*)((char*)d_ws + 256);
    float*        out     = (float*)d_out;

    berhu_init_kernel<<<1, 64, 0, stream>>>(ws_max);

    dim3 grid(BLOCKS_PER_SAMPLE, N_SAMPLES);
    berhu_max_kernel<<<grid, THREADS, 0, stream>>>(pred, targ, ws_max);
    berhu_sum_kernel<<<grid, THREADS, 0, stream>>>(pred, targ, ws_max, ws_part);
    berhu_final_kernel<<<1, 64, 0, stream>>>(ws_part, out);
}